// VarianceAdaptor_50345606644148
// MI455X (gfx1250) — compile-verified
//
#include <hip/hip_runtime.h>
#include <hip/hip_bf16.h>
#include <stdint.h>

#define B_      32
#define S_      1024
#define H_      384
#define MAXLEN_ 8192

typedef float v4f __attribute__((ext_vector_type(4)));
typedef int   v4i __attribute__((vector_size(16)));   // matches builtin's param type

#ifndef __has_builtin
#define __has_builtin(x) 0
#endif

#define AS3 __attribute__((address_space(3)))

// gfx1250 async store-from-LDS path (builtin presence confirmed by round-1 probe).
#if defined(__AMDGCN__) && \
    __has_builtin(__builtin_amdgcn_global_store_async_from_lds_b128) && \
    __has_builtin(__builtin_amdgcn_s_wait_asynccnt)
#define USE_ASYNC_ZERO 1
#else
#define USE_ASYNC_ZERO 0
#endif

// CPol: TH[2:0]; TH_STORE_NT = 1 (non-temporal streaming store)
#define CPOL_NT 1

// ---------------------------------------------------------------------------
// Kernel 3 (defined first so it leads the disasm snippet):
// one wave32 per frame. 384 floats = 3 x float4 per lane.
// Valid frames: b128 load from x row (L2-resident) -> NT b128 store.
// Zero frames:  async NT DMA from a 512B LDS zero block (gfx1250 path).
// ---------------------------------------------------------------------------
__global__ __launch_bounds__(256) void lr_gather(const float* __restrict__ x,
                                                 const int* __restrict__ idx,
                                                 float* __restrict__ out) {
#if USE_ASYNC_ZERO
  __shared__ __attribute__((aligned(64))) float zblk[128];   // 512B of zeros
  for (int i = threadIdx.x; i < 128; i += 256) zblk[i] = 0.0f;
  __syncthreads();
#endif
  const int lane = threadIdx.x & 31;
  const int w    = blockIdx.x * 8 + (threadIdx.x >> 5);      // frame id
  const int b    = w >> 13;
  const int t    = w & (MAXLEN_ - 1);
  const int sidx = idx[w];                                   // uniform per wave

  float* dst = out + ((size_t)b * MAXLEN_ + t) * (size_t)H_ + (size_t)lane * 4;

  if (sidx >= 0) {
    const float* src = x + ((size_t)b * S_ + (size_t)sidx) * (size_t)H_ +
                       (size_t)lane * 4;
    __builtin_prefetch(src, 0, 3);                 // global_prefetch_b8
#pragma unroll
    for (int k = 0; k < 3; ++k) {
      v4f v = *(const v4f*)(src + k * 128);        // global_load_b128
      __builtin_nontemporal_store(v, (v4f*)(dst + k * 128));  // th:NT b128
    }
  } else {
#if USE_ASYNC_ZERO
    // LDS source: 16B per lane from the shared zero block (LDS offset = low 32
    // bits of the generic LDS address, per the flat-aperture mapping).
    AS3 v4i* lsrc =
        (AS3 v4i*)(unsigned)((uintptr_t)&zblk[0] + (unsigned)lane * 16u);
#pragma unroll
    for (int k = 0; k < 3; ++k) {
      __builtin_amdgcn_global_store_async_from_lds_b128(
          (v4i*)(dst + k * 128), lsrc, 0, CPOL_NT);
    }
    __builtin_amdgcn_s_wait_asynccnt(0);           // drain before LDS dealloc
#else
    const v4f z = {0.0f, 0.0f, 0.0f, 0.0f};
#pragma unroll
    for (int k = 0; k < 3; ++k)
      __builtin_nontemporal_store(z, (v4f*)(dst + k * 128));
#endif
  }
}

// ---------------------------------------------------------------------------
// Kernel 1: per-batch inclusive scan of duration (LDS Hillis-Steele, 1024 thr)
// cum[b*S + i] = sum_{j<=i} dur[b][j];  mel_len_out[b] = (float)cum[b][S-1]
// ---------------------------------------------------------------------------
__global__ __launch_bounds__(S_) void lr_scan(const int* __restrict__ dur,
                                              int* __restrict__ cum,
                                              float* __restrict__ mel_len_out) {
  __shared__ int s[S_];
  const int b = blockIdx.x;
  const int i = threadIdx.x;
  s[i] = dur[b * S_ + i];
  __syncthreads();
#pragma unroll
  for (int off = 1; off < S_; off <<= 1) {
    int v = (i >= off) ? s[i - off] : 0;
    __syncthreads();
    s[i] += v;
    __syncthreads();
  }
  cum[b * S_ + i] = s[i];
  if (i == S_ - 1) mel_len_out[b] = (float)s[i];
}

// ---------------------------------------------------------------------------
// Kernel 2: one thread per output frame -> source index via upper_bound(cum,t)
// idx = searchsorted_right clamped to S-1, or -1 if t >= mel_len (zero frame)
// ---------------------------------------------------------------------------
__global__ __launch_bounds__(256) void lr_index(const int* __restrict__ cum,
                                                int* __restrict__ idx) {
  const int g = blockIdx.x * 256 + threadIdx.x;     // 0 .. B*MAXLEN-1
  const int b = g >> 13;                            // MAXLEN = 2^13
  const int t = g & (MAXLEN_ - 1);
  const int* c = cum + b * S_;
  const int mel_len = c[S_ - 1];
  int r = -1;
  if (t < mel_len) {
    int lo = 0, hi = S_;
#pragma unroll
    for (int it = 0; it < 10; ++it) {               // ceil(log2(1024)) steps
      int mid = (lo + hi) >> 1;
      if (c[mid] <= t) lo = mid + 1; else hi = mid;
    }
    r = (lo < S_ - 1) ? lo : (S_ - 1);
  }
  idx[g] = r;
}

// ---------------------------------------------------------------------------
// Kernel 4: mel_mask passthrough (bool bytes -> float)
// ---------------------------------------------------------------------------
__global__ __launch_bounds__(256) void lr_mask(const unsigned char* __restrict__ m,
                                               float* __restrict__ o) {
  const int g = blockIdx.x * 256 + threadIdx.x;    // 0 .. B*MAXLEN-1
  o[g] = m[g] ? 1.0f : 0.0f;
}

// ---------------------------------------------------------------------------
extern "C" void kernel_launch(void* const* d_in, const int* in_sizes, int n_in,
                              void* d_out, int out_size, void* d_ws, size_t ws_size,
                              hipStream_t stream) {
  (void)in_sizes; (void)n_in; (void)out_size; (void)ws_size;

  // setup_inputs order: x, src_mask, mel_mask, duration_target, pitch_target, max_len
  const float*         x        = (const float*)d_in[0];
  const unsigned char* mel_mask = (const unsigned char*)d_in[2];
  const int*           dur      = (const int*)d_in[3];

  float* out = (float*)d_out;
  // d_out layout (flat, return order): out | mel_len | mel_mask
  float* mel_len_out = out + (size_t)B_ * MAXLEN_ * H_;
  float* mask_out    = mel_len_out + B_;

  // workspace: cum (B*S ints) | idx (B*MAXLEN ints)  ~= 1.2 MB
  int* cum = (int*)d_ws;
  int* idx = cum + B_ * S_;

  lr_scan  <<<B_, S_, 0, stream>>>(dur, cum, mel_len_out);
  lr_index <<<(B_ * MAXLEN_) / 256, 256, 0, stream>>>(cum, idx);
  lr_gather<<<(B_ * MAXLEN_) / 8,   256, 0, stream>>>(x, idx, out);
  lr_mask  <<<(B_ * MAXLEN_) / 256, 256, 0, stream>>>(mel_mask, mask_out);
}